// TMessagePassing_927712936180
// MI455X (gfx1250) — compile-verified
//
#include <hip/hip_runtime.h>
#include <hip/hip_bf16.h>

// ---------------------------------------------------------------------------
// Hypergraph message passing, MI455X (gfx1250, wave32).
// One wave per output node; segmented reduction over sorted edge_owner.
// Gather-bound: 268 MB of embedding reads; emb table (25.6 MB) is L2-resident
// (192 MB L2). Member rows are staged into LDS with the gfx1250 async-to-LDS
// pipe (GLOBAL_LOAD_ASYNC_TO_LDS_B128, tracked by ASYNCcnt) and double-
// buffered so the next edge's 2 KB gather overlaps the current edge's math.
// ---------------------------------------------------------------------------

#define K_MEM 8
#define D_DIM 64
#define M_THRESH 4
#define WAVES_PER_BLOCK 8
#define ROW_BYTES (D_DIM * 4)              // 256 B per embedding row
#define EDGE_BYTES (K_MEM * ROW_BYTES)     // 2048 B staged per edge
#define SLOT_FLOATS (K_MEM * D_DIM)        // 512 floats per buffer
#define WAVE_FLOATS (2 * SLOT_FLOATS)      // double buffer per wave

__global__ __launch_bounds__(32 * WAVES_PER_BLOCK)
void hyperedge_mp_kernel(const int* __restrict__ nodes,        // [B]
                         const int* __restrict__ edge_owner,   // [E] sorted ascending
                         const int* __restrict__ edge_nodes,   // [E,K]
                         const int* __restrict__ edge_sizes,   // [E] in [2,K]
                         const float* __restrict__ emb,        // [N_EMB, 64]
                         const float* __restrict__ w_att_w,    // scalar
                         const float* __restrict__ w_att_b,    // scalar
                         float* __restrict__ out,              // [B, 64]
                         int B_, int E_)
{
    __shared__ __align__(16) float smem[WAVES_PER_BLOCK * WAVE_FLOATS]; // 32 KB

    const int lane = threadIdx.x & 31;                 // wave32
    const int wave = threadIdx.x >> 5;
    const int b    = blockIdx.x * WAVES_PER_BLOCK + wave;   // output node of this wave
    if (b >= B_) return;

    // wave-uniform copy of b so the binary search stays in SALU/SMEM
    const int bs = __builtin_amdgcn_readfirstlane(b);

    // ---- segment bounds: lower_bound(bs), lower_bound(bs+1) on sorted owners ----
    int lo = 0, hi = E_;
    while (lo < hi) { int mid = (lo + hi) >> 1; if (edge_owner[mid] <  bs) lo = mid + 1; else hi = mid; }
    const int start = lo;
    hi = E_;
    while (lo < hi) { int mid = (lo + hi) >> 1; if (edge_owner[mid] <= bs) lo = mid + 1; else hi = mid; }
    const int end = lo;

    const float ww = w_att_w[0];
    const float wb = w_att_b[0];

    // per-wave LDS slot (double buffered)
    const int      slot_f    = wave * WAVE_FLOATS;                       // float index
    const unsigned slot_byte = (unsigned)(size_t)(&smem[slot_f]);        // LDS byte addr

    // lane owns feature dims {2*lane, 2*lane+1}
    float2 acc; acc.x = 0.0f; acc.y = 0.0f;

    if (start < end) {
        // ---- prologue: stage edge `start` into buffer 0 ----
        int idxB  = edge_nodes[start * K_MEM + (lane & (K_MEM - 1))];
        int sizeA = edge_sizes[start];
        {
            // 4 async b128 ops cover 8 rows x 256B: lanes 0..15 -> member 2i,
            // lanes 16..31 -> member 2i+1, 16B per lane.
            #pragma unroll
            for (int i = 0; i < 4; ++i) {
                const int nk = __shfl(idxB, 2 * i + (lane >> 4), 32);
                const char* gaddr = (const char*)(emb + nk * D_DIM) + (lane & 15) * 16;
                const unsigned laddr = slot_byte + (unsigned)(i * 512 + lane * 16);
                asm volatile("global_load_async_to_lds_b128 %0, %1, off"
                             :: "v"(laddr), "v"(gaddr) : "memory");
            }
        }
        // metadata pipeline: idxB/sizeB describe edge e+1
        int sizeB = 0;
        if (start + 1 < end) {
            idxB  = edge_nodes[(start + 1) * K_MEM + (lane & (K_MEM - 1))];
            sizeB = edge_sizes[start + 1];
        }

        for (int e = start; e < end; ++e) {
            const int buf = (e - start) & 1;
            const int en  = e + 1;

            int idxC = 0, sizeC = 0;
            if (en < end) {
                // stage edge en into the other buffer while we compute edge e
                #pragma unroll
                for (int i = 0; i < 4; ++i) {
                    const int nk = __shfl(idxB, 2 * i + (lane >> 4), 32);
                    const char* gaddr = (const char*)(emb + nk * D_DIM) + (lane & 15) * 16;
                    const unsigned laddr = slot_byte +
                        (unsigned)((buf ^ 1) * EDGE_BYTES + i * 512 + lane * 16);
                    asm volatile("global_load_async_to_lds_b128 %0, %1, off"
                                 :: "v"(laddr), "v"(gaddr) : "memory");
                }
                if (en + 1 < end) {
                    idxC  = edge_nodes[(en + 1) * K_MEM + (lane & (K_MEM - 1))];
                    sizeC = edge_sizes[en + 1];
                }
                // async ops complete in order: <=4 outstanding means the
                // previous buffer (edge e) is fully in LDS.
                asm volatile("s_wait_asynccnt 0x4" ::: "memory");
            } else {
                asm volatile("s_wait_asynccnt 0x0" ::: "memory");
            }

            const int size = sizeA;                   // wave-uniform
            const float inv = 1.0f / (float)size;

            // consume staged rows: ds_load_b64 per member, bank-conflict-free
            float2 f[K_MEM];
            const int fbase = slot_f + buf * SLOT_FLOATS + lane * 2;
            #pragma unroll
            for (int k = 0; k < K_MEM; ++k)
                f[k] = *(const float2*)&smem[fbase + k * D_DIM];

            // masked mean over valid members
            float2 s; s.x = 0.0f; s.y = 0.0f;
            #pragma unroll
            for (int k = 0; k < K_MEM; ++k)
                if (k < size) { s.x += f[k].x; s.y += f[k].y; }
            float2 mean; mean.x = s.x * inv; mean.y = s.y * inv;

            // masked variance, exactly as reference: sum((x-mean)^2)/size, mean over D
            float v = 0.0f;
            #pragma unroll
            for (int k = 0; k < K_MEM; ++k)
                if (k < size) {
                    float dx = f[k].x - mean.x;
                    float dy = f[k].y - mean.y;
                    v += dx * dx + dy * dy;
                }
            // 32-lane reduction (wave32): 5 xor-shuffle steps
            #pragma unroll
            for (int off = 16; off > 0; off >>= 1) v += __shfl_xor(v, off, 32);
            const float edge_var = v * inv * (1.0f / (float)D_DIM);

            // 1x1 linear + sigmoid
            const float att = 1.0f / (1.0f + __expf(-(edge_var * ww + wb)));

            // product over first (size-1) valid members
            float2 msg;
            if (size == M_THRESH) {
                float2 p; p.x = 1.0f; p.y = 1.0f;
                #pragma unroll
                for (int k = 0; k < K_MEM; ++k)
                    if (k < size - 1) { p.x *= f[k].x; p.y *= f[k].y; }
                msg = p;
            } else {
                msg = mean;
            }

            acc.x += att * msg.x;
            acc.y += att * msg.y;

            // rotate metadata pipeline
            sizeA = sizeB;
            idxB  = idxC;
            sizeB = sizeC;
        }
    }

    float2 res;
    if (end > start) {
        res = acc;                                   // aggregated messages
    } else {
        const int n = nodes[b];                      // isolated node: self embedding
        res = ((const float2*)(emb + n * D_DIM))[lane];
    }
    ((float2*)(out + b * D_DIM))[lane] = res;
}

// ---------------------------------------------------------------------------
extern "C" void kernel_launch(void* const* d_in, const int* in_sizes, int n_in,
                              void* d_out, int out_size, void* d_ws, size_t ws_size,
                              hipStream_t stream)
{
    const int*   nodes      = (const int*)  d_in[0];   // [B]
    const int*   edge_owner = (const int*)  d_in[1];   // [E]
    const int*   edge_nodes = (const int*)  d_in[2];   // [E,K]
    const int*   edge_sizes = (const int*)  d_in[3];   // [E]
    const float* emb        = (const float*)d_in[4];   // [N_EMB, D]
    const float* w_att_w    = (const float*)d_in[5];   // scalar
    const float* w_att_b    = (const float*)d_in[6];   // scalar
    float*       out        = (float*)d_out;

    const int B_ = in_sizes[0];
    const int E_ = in_sizes[1];

    const int threads = 32 * WAVES_PER_BLOCK;
    const int blocks  = (B_ + WAVES_PER_BLOCK - 1) / WAVES_PER_BLOCK;

    hyperedge_mp_kernel<<<blocks, threads, 0, stream>>>(
        nodes, edge_owner, edge_nodes, edge_sizes, emb, w_att_w, w_att_b,
        out, B_, E_);

    (void)n_in; (void)out_size; (void)d_ws; (void)ws_size;
}